// Attention_78537771975200
// MI455X (gfx1250) — compile-verified
//
#include <hip/hip_runtime.h>

// ---- problem constants ----
#define BN   16      // bs*N
#define T_   12
#define TL   11      // T-1 attention steps
#define TP   9       // T - N_TS + 1
#define CH   128
#define HW   576
#define WD   24
#define CVP  128     // padded V rows (96 proj + 18 PE + 14 pad)
#define CB   114     // valid channels per output block
#define CO   456     // output channels
#define EPSN 1e-12f
#define SCALE 0.08838834764831845f  // 128^-0.5

typedef __attribute__((ext_vector_type(2))) float        v2f;
typedef __attribute__((ext_vector_type(8))) float        v8f;
typedef __attribute__((ext_vector_type(4))) unsigned int v4u;
typedef __attribute__((ext_vector_type(8))) int          v8i;
typedef __attribute__((ext_vector_type(4))) int          v4i;

#define USE_TDM 0
#if defined(__has_builtin)
#if __has_builtin(__builtin_amdgcn_tensor_load_to_lds) && \
    __has_builtin(__builtin_amdgcn_s_wait_tensorcnt)
#undef USE_TDM
#define USE_TDM 1
#endif
#endif

__device__ __forceinline__ v8f wmma_f32(v2f a, v2f b, v8f c) {
  // V_WMMA_F32_16X16X4_F32 : D = A(16x4 f32) x B(4x16 f32) + C(16x16 f32)
  return __builtin_amdgcn_wmma_f32_16x16x4_f32(false, a, false, b, (short)0, c,
                                               false, false);
}

// ---------- K1: per-pixel channel L2 normalize (one batch) ----------
__global__ void k_normalize(const float* __restrict__ x_b, float* __restrict__ xn_b) {
  int g = blockIdx.x * blockDim.x + threadIdx.x;   // over T_*HW pixels
  if (g >= T_ * HW) return;
  int t = g / HW, n = g % HW;
  const float* src = x_b + (size_t)t * CH * HW + n;
  float ss = 0.f;
  for (int c = 0; c < CH; ++c) { float v = src[(size_t)c * HW]; ss += v * v; }
  float inv = 1.f / fmaxf(sqrtf(ss), EPSN);
  float* dst = xn_b + (size_t)t * CH * HW + n;
  for (int c = 0; c < CH; ++c) dst[(size_t)c * HW] = src[(size_t)c * HW] * inv;
}

// ---------- fill PE rows 96..113 of Vp (same for all b, filled once/launch) ----------
__global__ void k_fill_pe(float* __restrict__ vp_b) {
  int g = blockIdx.x * blockDim.x + threadIdx.x;   // T_*18*HW
  if (g >= T_ * 18 * HW) return;
  int n = g % HW; int rest = g / HW; int p = rest % 18; int t = rest / 18;
  int i = n / WD, j = n % WD;
  float y  = -1.f + 2.f * (float)i / (float)(WD - 1);
  float xq = -1.f + 2.f * (float)j / (float)(WD - 1);
  float val;
  if (p == 0)      val = y;
  else if (p == 1) val = xq;
  else {
    int q = p - 2; int fi = q >> 2; int g4 = q & 3;
    float f = (float)(1 << fi) * 3.14159265358979323846f;
    float coord = (g4 & 1) ? xq : y;
    val = (g4 < 2) ? sinf(f * coord) : cosf(f * coord);
  }
  vp_b[((size_t)t * CVP + 96 + p) * HW + n] = val;
}

// ---------- K2: Vp rows 0..95 = Wv[32+o,:] @ x[t] + bv  (WMMA fp32) ----------
__global__ void k_project(const float* __restrict__ x_b, const float* __restrict__ Wv,
                          const float* __restrict__ bv, float* __restrict__ vp_b) {
  int nt = blockIdx.x;          // 0..35  n tile
  int rt = blockIdx.y;          // 0..5   row tile (out ch 32+16*rt ..)
  int t  = blockIdx.z;          // 0..11
  int L  = threadIdx.x;         // one wave32
  int lm = L & 15;
  int koff = (L >> 4) << 1;     // A/B K split across half-waves
  int orow = 32 + rt * 16 + lm;
  int n0 = nt * 16;
  const float* xb = x_b + (size_t)t * CH * HW;
  v8f acc = {};
  for (int k = 0; k < CH; k += 4) {
    v2f a, b;
    a.x = Wv[(size_t)orow * CH + k + koff];
    a.y = Wv[(size_t)orow * CH + k + koff + 1];
    b.x = xb[(size_t)(k + koff) * HW + n0 + lm];
    b.y = xb[(size_t)(k + koff + 1) * HW + n0 + lm];
    acc = wmma_f32(a, b, acc);
  }
  int mb = (L >> 4) * 8;
  float* dst = vp_b + (size_t)t * CVP * HW;
  for (int r = 0; r < 8; ++r) {
    int row = rt * 16 + mb + r;                 // 0..95
    dst[(size_t)row * HW + n0 + lm] = acc[r] + bv[32 + row];
  }
}

// ---------- K3: energy E[j][n][m] = scale/temp * sum_c q[c,n] k[c,m] ----------
__global__ void k_energy(const float* __restrict__ xn_b, float* __restrict__ E_b,
                         const int* __restrict__ temp_p) {
  int nt = blockIdx.x;          // 0..35  (16 n)
  int ms = blockIdx.y;          // 0..8   (64 m, 4 accumulators)
  int j  = blockIdx.z;          // 0..10
  int L = threadIdx.x; int lm = L & 15; int koff = (L >> 4) << 1;
  int n0 = nt * 16, m0 = ms * 64;
  const float* q  = xn_b + (size_t)(j + 1) * CH * HW;
  const float* km = xn_b + (size_t)j * CH * HW;
  v8f a0 = {}, a1 = {}, a2 = {}, a3 = {};
  for (int c = 0; c < CH; c += 4) {
    v2f a;
    a.x = q[(size_t)(c + koff) * HW + n0 + lm];
    a.y = q[(size_t)(c + koff + 1) * HW + n0 + lm];
    const float* kr0 = km + (size_t)(c + koff) * HW;
    const float* kr1 = km + (size_t)(c + koff + 1) * HW;
    v2f b0, b1, b2, b3;
    b0.x = kr0[m0 + lm];      b0.y = kr1[m0 + lm];
    b1.x = kr0[m0 + 16 + lm]; b1.y = kr1[m0 + 16 + lm];
    b2.x = kr0[m0 + 32 + lm]; b2.y = kr1[m0 + 32 + lm];
    b3.x = kr0[m0 + 48 + lm]; b3.y = kr1[m0 + 48 + lm];
    a0 = wmma_f32(a, b0, a0); a1 = wmma_f32(a, b1, a1);
    a2 = wmma_f32(a, b2, a2); a3 = wmma_f32(a, b3, a3);
  }
  float sc = SCALE / (float)temp_p[0];
  float* e = E_b + (size_t)j * HW * HW;
  int mb = (L >> 4) * 8;
  for (int r = 0; r < 8; ++r) {
    float* erow = e + (size_t)(n0 + mb + r) * HW;
    erow[m0 + lm]      = a0[r] * sc;
    erow[m0 + 16 + lm] = a1[r] * sc;
    erow[m0 + 32 + lm] = a2[r] * sc;
    erow[m0 + 48 + lm] = a3[r] * sc;
  }
}

// ---------- K4: softmax over m (rows of E) + transposed store At[m][n] ----------
__global__ void k_softmax_tr(const float* __restrict__ E_b, float* __restrict__ At_b) {
  __shared__ float lds[16 * 577];                 // 16 rows, padded (bank-safe)
  int nt = blockIdx.x;                            // 0..35 (16 n-rows each)
  int j  = blockIdx.y;
  int tid = threadIdx.x;                          // 512 threads = 16 waves
  const float* e = E_b + (size_t)j * HW * HW + (size_t)nt * 16 * HW;
  for (int idx = tid; idx < 16 * HW; idx += 512) {
    int row = idx / HW, m = idx % HW;
    lds[row * 577 + m] = e[(size_t)row * HW + m];
  }
  __syncthreads();
  int w = tid >> 5, lane = tid & 31;              // wave w owns row w
  float* r = lds + w * 577;
  float vmax = -3.4e38f;
  for (int m = lane; m < HW; m += 32) vmax = fmaxf(vmax, r[m]);
  for (int off = 16; off >= 1; off >>= 1) vmax = fmaxf(vmax, __shfl_xor(vmax, off, 32));
  float ssum = 0.f;
  for (int m = lane; m < HW; m += 32) { float ev = __expf(r[m] - vmax); r[m] = ev; ssum += ev; }
  for (int off = 16; off >= 1; off >>= 1) ssum += __shfl_xor(ssum, off, 32);
  float inv = 1.f / ssum;
  for (int m = lane; m < HW; m += 32) r[m] *= inv;
  __syncthreads();
  float* at = At_b + (size_t)j * HW * HW + nt * 16;
  for (int idx = tid; idx < 16 * HW; idx += 512) {
    int nl = idx & 15, m = idx >> 4;
    at[(size_t)m * HW + nl] = lds[nl * 577 + m];  // coalesced 64B segments
  }
}

// ---------- K5: Y[j][c][n] = sum_m X[j][c][m] * At[j][m][n]  (row chain GEMM) ----------
// A-tile (16 c x 64 m, row stride 576 f32) staged into LDS with stride 66 via the
// Tensor Data Mover (pad 2 dwords after every 64 dwords), synced by s_wait_tensorcnt.
__global__ void k_rowmm(const float* __restrict__ X, size_t Xjs,
                        const float* __restrict__ At, size_t Ats,
                        float* __restrict__ Y, size_t Yjs, int rows_valid) {
  __shared__ float ldsX[16 * 66];                 // only static LDS -> offset 0
  int ns = blockIdx.x;                            // 0..8  (64 n per block)
  int ct = blockIdx.y;                            // 0..7  (16 c)
  int j  = blockIdx.z;
  int tid = threadIdx.x;                          // 128 threads = 4 waves
  int wv = tid >> 5, L = tid & 31, lm = L & 15, koff = (L >> 4) << 1;
  int c0 = ct * 16;
  int n0 = ns * 64 + wv * 16;                     // each wave owns one n tile
  const float* Xj = X + (size_t)j * Xjs + (size_t)c0 * HW;
  const float* Aj = At + (size_t)j * Ats;
  v8f acc = {};
  for (int kb = 0; kb < HW; kb += 64) {
    __syncthreads();                              // WAR: LDS readers done
#if USE_TDM
    if (tid < 32) {                               // wave 0 issues the tile DMA
      unsigned long long ga = (unsigned long long)(const void*)(Xj + kb);
      v4u g0;
      g0[0] = 1u;                                 // count=1 (valid user D#)
      g0[1] = 0u;                                 // lds_addr = 0 (ldsX)
      g0[2] = (unsigned)(ga & 0xFFFFFFFFu);       // global_addr[31:0]
      g0[3] = (unsigned)((ga >> 32) & 0x01FFFFFFu) | (2u << 30);  // [56:32] | type=2
      v8i g1;
      g1[0] = (int)((2u << 16)      // data_size = 4B
                  | (1u << 20)      // pad_enable
                  | (5u << 22)      // pad_interval: 64 dwords
                  | (1u << 25));    // pad_amount : 2 dwords -> LDS row stride 66
      g1[1] = (int)((576u & 0xFFFFu) << 16);      // tensor_dim0[15:0]
      g1[2] = (int)(((576u >> 16) & 0xFFFFu) | ((128u & 0xFFFFu) << 16)); // td0 hi | td1 lo
      g1[3] = (int)(((128u >> 16) & 0xFFFFu) | (64u << 16));  // td1 hi | tile_dim0=64
      g1[4] = (int)(16u);                         // tile_dim1=16, tile_dim2=0
      g1[5] = (int)(576u);                        // tensor_dim0_stride[31:0]
      g1[6] = 0;                                  // stride hi | dim1_stride lo
      g1[7] = 0;
      v4i gz4 = {0, 0, 0, 0};
      v8i gz8 = {0, 0, 0, 0, 0, 0, 0, 0};
      __builtin_amdgcn_tensor_load_to_lds(g0, g1, gz4, gz4, gz8, 0);
      __builtin_amdgcn_s_wait_tensorcnt(0);       // drain TENSORcnt
    }
#else
    for (int idx = tid; idx < 16 * 64; idx += 128) {
      int row = idx >> 6, kk = idx & 63;
      ldsX[row * 66 + kk] = Xj[(size_t)row * HW + kb + kk];
    }
#endif
    __syncthreads();                              // RAW: tile visible to all waves
    if (kb + 64 < HW)                             // CDNA5 global_prefetch_b8
      __builtin_prefetch(&Aj[(size_t)(kb + 64) * HW + n0 + lm], 0, 1);
    for (int kk = 0; kk < 64; kk += 4) {
      v2f a, b;
      a.x = ldsX[lm * 66 + kk + koff];
      a.y = ldsX[lm * 66 + kk + koff + 1];
      int m = kb + kk + koff;
      b.x = Aj[(size_t)m * HW + n0 + lm];
      b.y = Aj[(size_t)(m + 1) * HW + n0 + lm];
      acc = wmma_f32(a, b, acc);
    }
  }
  float* Yj = Y + (size_t)j * Yjs;
  int mb = (L >> 4) * 8;
  for (int r = 0; r < 8; ++r) {
    int row = c0 + mb + r;
    if (row < rows_valid) Yj[(size_t)row * HW + n0 + lm] = acc[r];
  }
}

// ---------- K6: gather output blocks 0..2 (block 3 written by stage-3 rowmm) ----------
__global__ void k_assemble(const float* __restrict__ vp_b, const float* __restrict__ W1_b,
                           const float* __restrict__ W2_b, float* __restrict__ out_b) {
  int g = blockIdx.x * blockDim.x + threadIdx.x;
  const int TOT = TP * 342 * HW;
  if (g >= TOT) return;
  int n = g % HW; int rest = g / HW; int c = rest % 342; int s = rest / 342;
  float v;
  if (c < CB)        v = vp_b[((size_t)(s + 3) * CVP + c) * HW + n];
  else if (c < 228)  v = W1_b[((size_t)(s + 2) * CVP + (c - CB)) * HW + n];
  else               v = W2_b[((size_t)(s + 1) * CVP + (c - 228)) * HW + n];
  out_b[((size_t)s * CO + c) * HW + n] = v;
}

extern "C" void kernel_launch(void* const* d_in, const int* in_sizes, int n_in,
                              void* d_out, int out_size, void* d_ws, size_t ws_size,
                              hipStream_t stream) {
  (void)in_sizes; (void)n_in; (void)out_size; (void)ws_size;
  const float* x    = (const float*)d_in[0];
  const float* Wv   = (const float*)d_in[1];
  const float* bv   = (const float*)d_in[2];
  const int*   temp = (const int*)d_in[3];
  float* out = (float*)d_out;

  // per-batch scratch (reused across b; ~42.5 MB -> L2 resident)
  float* xn = (float*)d_ws;                       // T*CH*HW      = 884736
  float* vp = xn + (size_t)T_ * CH * HW;          // T*CVP*HW     = 884736
  float* E  = vp + (size_t)T_ * CVP * HW;         // TL*HW*HW     = 3649536
  float* At = E  + (size_t)TL * HW * HW;          // TL*HW*HW     = 3649536
  float* W1 = At + (size_t)TL * HW * HW;          // TL*CVP*HW    = 811008
  float* W2 = W1 + (size_t)TL * CVP * HW;         // 10*CVP*HW    = 737280

  k_fill_pe<<<dim3((T_ * 18 * HW + 255) / 256), 256, 0, stream>>>(vp);

  for (int b = 0; b < BN; ++b) {
    const float* xb = x + (size_t)b * T_ * CH * HW;
    float* outb = out + (size_t)b * TP * CO * HW;

    k_normalize<<<dim3((T_ * HW + 255) / 256), 256, 0, stream>>>(xb, xn);
    k_project  <<<dim3(36, 6, T_), 32, 0, stream>>>(xb, Wv, bv, vp);
    k_energy   <<<dim3(36, 9, TL), 32, 0, stream>>>(xn, E, temp);
    k_softmax_tr<<<dim3(36, TL), 512, 0, stream>>>(E, At);

    // W1[j] = V[j] @ A[j]^T        j = 0..10
    k_rowmm<<<dim3(9, 8, TL), 128, 0, stream>>>(
        vp, (size_t)CVP * HW, At, (size_t)HW * HW, W1, (size_t)CVP * HW, CVP);
    // W2[j] = W1[j] @ A[j+1]^T     j = 0..9
    k_rowmm<<<dim3(9, 8, 10), 128, 0, stream>>>(
        W1, (size_t)CVP * HW, At + (size_t)HW * HW, (size_t)HW * HW,
        W2, (size_t)CVP * HW, CVP);
    // block3: out[s=j, 342:456] = W2[j] @ A[j+2]^T   j = 0..8 (direct store)
    k_rowmm<<<dim3(9, 8, TP), 128, 0, stream>>>(
        W2, (size_t)CVP * HW, At + (size_t)2 * HW * HW, (size_t)HW * HW,
        outb + (size_t)342 * HW, (size_t)CO * HW, CB);

    k_assemble<<<dim3((TP * 342 * HW + 255) / 256), 256, 0, stream>>>(vp, W1, W2, outb);
  }
}